// GeGate_82214263980323
// MI455X (gfx1250) — compile-verified
//
#include <hip/hip_runtime.h>

// Elementwise: out = capped_relu(x * w + b), x:[batch, d] f32, w/b:[d] f32.
// Memory-bound (0.25 FLOP/B). Strategy: float4 (b128) NT streaming of x/out,
// w4/b4 held in registers per thread across a 16-row strip.

typedef float float4v __attribute__((ext_vector_type(4)));

__global__ __launch_bounds__(256) void capped_relu_diag_kernel(
    const float* __restrict__ x,
    const float* __restrict__ w,
    const float* __restrict__ b,
    float* __restrict__ out,
    int batch, int cols4) {
  const int c4 = blockIdx.x * blockDim.x + threadIdx.x;
  if (c4 >= cols4) return;

  const float4v* __restrict__ x4 = (const float4v*)x;
  float4v* __restrict__ o4 = (float4v*)out;

  // Per-thread column parameters: loaded once, reused for every row this
  // thread touches (regular cached loads; w/b are 32KB total -> L2 resident).
  const float4v wv = ((const float4v*)w)[c4];
  const float4v bv = ((const float4v*)b)[c4];

  const long rowStride = (long)cols4;          // float4 units per row
  const int rstep = gridDim.y;

  int row = blockIdx.y;
  // Main unrolled-by-4 strip: 4 independent b128 NT loads in flight.
  for (; row + 3 * rstep < batch; row += 4 * rstep) {
#pragma unroll
    for (int u = 0; u < 4; ++u) {
      const long idx = ((long)row + (long)u * rstep) * rowStride + c4;
      // Prefetch one unroll-group ahead (speculative; OOB is dropped).
      __builtin_prefetch(&x4[idx + 4L * rstep * rowStride], 0, 0);
      float4v xv = __builtin_nontemporal_load(&x4[idx]);
      float4v y = xv * wv + bv;
      float4v r;
#pragma unroll
      for (int i = 0; i < 4; ++i) {
        float yi = y[i];
        r[i] = (yi > 0.0f && yi <= 1.0f) ? yi : 0.0f;
      }
      __builtin_nontemporal_store(r, &o4[idx]);
    }
  }
  // Tail rows.
  for (; row < batch; row += rstep) {
    const long idx = (long)row * rowStride + c4;
    float4v xv = __builtin_nontemporal_load(&x4[idx]);
    float4v y = xv * wv + bv;
    float4v r;
#pragma unroll
    for (int i = 0; i < 4; ++i) {
      float yi = y[i];
      r[i] = (yi > 0.0f && yi <= 1.0f) ? yi : 0.0f;
    }
    __builtin_nontemporal_store(r, &o4[idx]);
  }
}

// Scalar fallback for d % 4 != 0 (not hit for d=4096, kept for generality).
__global__ __launch_bounds__(256) void capped_relu_diag_scalar(
    const float* __restrict__ x,
    const float* __restrict__ w,
    const float* __restrict__ b,
    float* __restrict__ out,
    long n, int d) {
  long i = (long)blockIdx.x * blockDim.x + threadIdx.x;
  const long stride = (long)gridDim.x * blockDim.x;
  for (; i < n; i += stride) {
    int c = (int)(i % d);
    float y = x[i] * w[c] + b[c];
    out[i] = (y > 0.0f && y <= 1.0f) ? y : 0.0f;
  }
}

extern "C" void kernel_launch(void* const* d_in, const int* in_sizes, int n_in,
                              void* d_out, int out_size, void* d_ws, size_t ws_size,
                              hipStream_t stream) {
  const float* x = (const float*)d_in[0];
  const float* w = (const float*)d_in[1];
  const float* b = (const float*)d_in[2];
  float* out = (float*)d_out;

  const long n = (long)in_sizes[0];   // batch * d
  const int d = in_sizes[1];          // feature dim (4096)
  const int batch = (int)(n / d);

  if ((d & 3) == 0) {
    const int cols4 = d >> 2;                       // 1024 float4 per row
    dim3 block(256);
    int gx = (cols4 + 255) / 256;                   // 4 column blocks
    int gy = batch < 512 ? batch : 512;             // 16 rows per block strip
    dim3 grid(gx, gy);
    capped_relu_diag_kernel<<<grid, block, 0, stream>>>(x, w, b, out, batch, cols4);
  } else {
    int blocks = (int)((n + 255) / 256);
    if (blocks > 16384) blocks = 16384;
    capped_relu_diag_scalar<<<blocks, 256, 0, stream>>>(x, w, b, out, n, d);
  }
}